// MaskedCrossEntropyLoss_90649579749773
// MI455X (gfx1250) — compile-verified
//
#include <hip/hip_runtime.h>
#include <hip/hip_bf16.h>

#define WAVE 32
#define BLOCK 256
#define NW (BLOCK / WAVE)

// ---------- CDNA5 async global->LDS copy (ASYNCcnt path) ----------
#if __has_builtin(__builtin_amdgcn_global_load_async_to_lds_b128)
#define HAVE_ASYNC_BUILTIN 1
#else
#define HAVE_ASYNC_BUILTIN 0
#warning "probe: __builtin_amdgcn_global_load_async_to_lds_b128 NOT available -> inline-asm async path"
#endif
#if __has_builtin(__builtin_amdgcn_s_wait_asynccnt)
#define HAVE_WAIT_BUILTIN 1
#else
#define HAVE_WAIT_BUILTIN 0
#warning "probe: __builtin_amdgcn_s_wait_asynccnt NOT available -> inline-asm s_wait_asynccnt"
#endif

typedef int v4i_t __attribute__((vector_size(4 * sizeof(int))));
typedef __attribute__((address_space(1))) v4i_t* gv4i_p;   // global (AS1)
typedef __attribute__((address_space(3))) v4i_t* lv4i_p;   // LDS (AS3)
typedef __attribute__((address_space(3))) void*  as3_vp;

__device__ __forceinline__ void async_ld_b128(const float* g, float* l) {
#if HAVE_ASYNC_BUILTIN
    __builtin_amdgcn_global_load_async_to_lds_b128((gv4i_p)g, (lv4i_p)l, 0, 0);
#else
    unsigned int loff = (unsigned int)(unsigned long long)(as3_vp)l;
    unsigned long long ga = (unsigned long long)g;
    asm volatile("global_load_async_to_lds_b128 %0, %1, off"
                 :: "v"(loff), "v"(ga) : "memory");
#endif
}

__device__ __forceinline__ void wait_async_grp32() {   // ASYNCcnt <= 32
#if HAVE_WAIT_BUILTIN
    __builtin_amdgcn_s_wait_asynccnt(32);
#else
    asm volatile("s_wait_asynccnt 0x20" ::: "memory");
#endif
    asm volatile("" ::: "memory");
}
__device__ __forceinline__ void wait_async_all() {     // ASYNCcnt <= 0
#if HAVE_WAIT_BUILTIN
    __builtin_amdgcn_s_wait_asynccnt(0);
#else
    asm volatile("s_wait_asynccnt 0x0" ::: "memory");
#endif
    asm volatile("" ::: "memory");
}

// -------------------------------------------------------------------
// Never launched: minimal body placed FIRST in the module so the device-asm
// snippet window shows the exact async mnemonics this toolchain emits.
__global__ void aaa_probe_async(const float* __restrict__ g, float* __restrict__ out) {
    __shared__ __align__(16) float buf[WAVE * 4];
    async_ld_b128(g + threadIdx.x * 4, buf + threadIdx.x * 4);
    wait_async_all();
    out[threadIdx.x] = buf[threadIdx.x * 4];
}

// d_ws layout: [0] double loss_sum, [1] u64 valid_count
__global__ void mce_init_ws(double* ws) {
    if (blockIdx.x == 0 && threadIdx.x == 0) {
        ws[0] = 0.0;
        ((unsigned long long*)ws)[1] = 0ull;
    }
}

// Issue async copies of one batch (up to 32 rows) into an LDS buffer.
// Row i of the batch lands at buf + i*rowFloats (lane-per-row consumption).
__device__ __forceinline__ int issue_batch(const float* __restrict__ logits,
                                           int C, int chunks, int batch,
                                           float* buf, int lane, int N,
                                           int rowFloats) {
    long long row0 = (long long)batch << 5;
    int n = (int)((long long)N - row0);
    if (n > 32) n = 32;
    int instr = 0;
    for (int i = 0; i < n; ++i) {
        const float* g = logits + (size_t)(row0 + i) * (size_t)C;
        float*       l = buf + (size_t)i * (size_t)rowFloats;
        for (int c = lane; c < chunks; c += WAVE)
            async_ld_b128(g + (size_t)c * 4, l + (size_t)c * 4);
        instr += (chunks + WAVE - 1) / WAVE;
    }
    return instr;
}

__global__ __launch_bounds__(BLOCK) void mce_main(
    const float* __restrict__ logits,
    const int*   __restrict__ targets,
    const int*   __restrict__ cls,
    int N, int C, int K,
    double* __restrict__ ws)
{
    extern __shared__ __align__(16) char smem[];
    const int tid  = threadIdx.x;
    const int lane = tid & (WAVE - 1);
    const int w    = tid >> 5;

    const int chunks    = C >> 2;          // 16B chunks per row (C % 4 == 0)
    const int rowFloats = C + 4;           // padded stride (16B aligned)
    const int bufFloats = 32 * rowFloats;  // one 32-row batch buffer

    float* buf0 = (float*)smem + (size_t)w * 2 * (size_t)bufFloats;
    float* buf1 = buf0 + bufFloats;
    int*   sel  = (int*)(smem + (size_t)NW * 2 * (size_t)bufFloats * 4);
    int*   cols = sel + C;
    float* wsum = (float*)(cols + K);
    int*   wcnt = (int*)(wsum + NW);

    for (int c = tid; c < C; c += BLOCK) sel[c] = 0;
    __syncthreads();
    for (int k = tid; k < K; k += BLOCK) {
        int c = cls[k];
        if (c >= 0 && c < C) sel[c] = 1;
        cols[k] = (c < 0) ? 0 : (c >= C ? C - 1 : c);
    }
    __syncthreads();

    const int wid = blockIdx.x * NW + w;
    const int nw  = gridDim.x * NW;
    const int numBatches = (N + 31) >> 5;
    const int fullInstr  = 32 * ((chunks + WAVE - 1) / WAVE);

    float acc = 0.0f;
    int   cnt = 0;

    if (wid < numBatches)
        (void)issue_batch(logits, C, chunks, wid, buf0, lane, N, rowFloats);

    int parity = 0;
    for (int b = wid; b < numBatches; b += nw) {
        float* nextBuf = parity ? buf0 : buf1;
        float* curBuf  = parity ? buf1 : buf0;

        int nb = b + nw;
        int issued_next = 0;
        if (nb < numBatches)
            issued_next = issue_batch(logits, C, chunks, nb, nextBuf, lane, N, rowFloats);

        // retire current batch; keep the 32 just-issued ops in flight
        if (issued_next == fullInstr && fullInstr == 32) wait_async_grp32();
        else                                             wait_async_all();

        long long row0 = (long long)b << 5;
        int n = (int)((long long)N - row0);
        if (n > 32) n = 32;

        if (lane < n) {
            const float* my = curBuf + (size_t)lane * (size_t)rowFloats;
            float mx = -3.0e38f;
            #pragma unroll 4
            for (int k = 0; k < K; ++k) mx = fmaxf(mx, my[cols[k]]);
            float s = 0.0f;
            #pragma unroll 4
            for (int k = 0; k < K; ++k) s += __expf(my[cols[k]] - mx);
            int t = targets[row0 + lane];
            bool valid = (t >= 0) && (t < C) && (sel[t] != 0);
            if (valid) {
                acc += (mx + __logf(s)) - my[t];
                cnt += 1;
            }
        }
        parity ^= 1;
    }

    // wave32 reduce, then block reduce, then one atomic pair per block
    #pragma unroll
    for (int off = 16; off > 0; off >>= 1) {
        acc += __shfl_xor(acc, off, WAVE);
        cnt += __shfl_xor(cnt, off, WAVE);
    }
    if (lane == 0) { wsum[w] = acc; wcnt[w] = cnt; }
    __syncthreads();
    if (tid == 0) {
        double bs = 0.0; long long bc = 0;
        #pragma unroll
        for (int i = 0; i < NW; ++i) { bs += (double)wsum[i]; bc += wcnt[i]; }
        unsafeAtomicAdd(&ws[0], bs);                                    // global_atomic_add_f64
        atomicAdd((unsigned long long*)ws + 1, (unsigned long long)bc);
    }
}

__global__ void mce_finalize(const double* __restrict__ ws, float* __restrict__ out) {
    if (blockIdx.x == 0 && threadIdx.x == 0) {
        double s = ws[0];
        double c = (double)((const unsigned long long*)ws)[1];
        out[0] = (float)(s / fmax(c, 1.0));
    }
}

extern "C" void kernel_launch(void* const* d_in, const int* in_sizes, int n_in,
                              void* d_out, int out_size, void* d_ws, size_t ws_size,
                              hipStream_t stream) {
    const float* logits  = (const float*)d_in[0];
    const int*   targets = (const int*)d_in[1];
    const int*   cls     = (const int*)d_in[2];
    float*  out = (float*)d_out;
    double* ws  = (double*)d_ws;

    const int N = in_sizes[1];
    const int C = (N > 0) ? (in_sizes[0] / N) : 0;
    const int K = in_sizes[2];

    const int rowFloats = C + 4;
    size_t shbytes = (size_t)NW * 2 * 32 * (size_t)rowFloats * 4   // wave double-buffers
                   + (size_t)C * 4 + (size_t)K * 4                 // sel LUT + cols
                   + (size_t)NW * 8;                               // wsum + wcnt
    (void)hipFuncSetAttribute((const void*)mce_main,
                              hipFuncAttributeMaxDynamicSharedMemorySize,
                              (int)shbytes);

    mce_init_ws<<<1, 32, 0, stream>>>(ws);

    int numBatches = (N + 31) / 32;
    int grid = 256;
    int maxGrid = (numBatches + NW - 1) / NW;
    if (grid > maxGrid) grid = (maxGrid > 0) ? maxGrid : 1;

    mce_main<<<grid, BLOCK, shbytes, stream>>>(logits, targets, cls, N, C, K, ws);
    mce_finalize<<<1, 32, 0, stream>>>(ws, out);
}